// OneScaleOneAttentionLayer_79156247265573
// MI455X (gfx1250) — compile-verified
//
#include <hip/hip_runtime.h>
#include <hip/hip_bf16.h>
#include <math.h>

#define B_   32
#define LQ_  1024
#define LA_  1024
#define H_   512
#define NEG_INF_ (-1.0e7f)

typedef float v2f __attribute__((ext_vector_type(2)));
typedef float v8f __attribute__((ext_vector_type(8)));

// D = A(16x4 f32) * B(4x16 f32) + C(16x16 f32)
__device__ __forceinline__ v8f wmma_f32x4(v2f a, v2f b, v8f c) {
    return __builtin_amdgcn_wmma_f32_16x16x4_f32(
        /*neg_a=*/false, a, /*neg_b=*/false, b,
        /*c_mod=*/(short)0, c, /*reuse_a=*/false, /*reuse_b=*/false);
}

// ---------------------------------------------------------------------------
// Kernel 1: S[b,q,a] = mask ? (q . a) * temperature : -1e7
// 128x128 output tile per workgroup, K staged in 64-wide LDS chunks.
// ---------------------------------------------------------------------------
__global__ __launch_bounds__(256) void score_kernel(
    const float* __restrict__ q, const float* __restrict__ a,
    const int* __restrict__ qmask, const int* __restrict__ amask,
    const float* __restrict__ temp, float* __restrict__ S)
{
    const int b     = blockIdx.z;
    const int qBase = blockIdx.y * 128;
    const int aBase = blockIdx.x * 128;
    const int tid   = threadIdx.x;
    const int lane  = tid & 31;
    const int wave  = tid >> 5;
    const int wm    = wave & 3;   // 4 M strips of 32 rows
    const int wn    = wave >> 2;  // 2 N strips of 64 cols
    const float T   = temp[0];

    __shared__ __align__(16) float qs[128 * 68];
    __shared__ __align__(16) float as[128 * 68];

    const float* qg = q + ((size_t)b * LQ_ + qBase) * H_;
    const float* ag = a + ((size_t)b * LA_ + aBase) * H_;

    v8f acc[2][4];
    #pragma unroll
    for (int i = 0; i < 2; ++i)
        #pragma unroll
        for (int j = 0; j < 4; ++j) acc[i][j] = {};

    for (int k0 = 0; k0 < H_; k0 += 64) {
        __syncthreads();
        // stage 128x64 tiles of q and a into LDS (float4, padded stride 68)
        #pragma unroll
        for (int i = 0; i < 8; ++i) {
            int idx = tid + i * 256;          // 0..2047
            int r   = idx >> 4;               // row 0..127
            int c4  = idx & 15;               // float4 col 0..15
            float4 vq = *(const float4*)(qg + (size_t)r * H_ + k0 + c4 * 4);
            *(float4*)(qs + r * 68 + c4 * 4) = vq;
            float4 va = *(const float4*)(ag + (size_t)r * H_ + k0 + c4 * 4);
            *(float4*)(as + r * 68 + c4 * 4) = va;
        }
        __syncthreads();

        const int khalf = (lane >> 4) * 2;    // 0 or 2 (lane-half K select)
        const int ml    = lane & 15;
        #pragma unroll
        for (int kk = 0; kk < 64; kk += 4) {
            v2f A[2], Bv[4];
            #pragma unroll
            for (int i = 0; i < 2; ++i) {
                int m = wm * 32 + i * 16 + ml;
                A[i] = *(const v2f*)(qs + m * 68 + kk + khalf);
            }
            #pragma unroll
            for (int j = 0; j < 4; ++j) {
                int n = wn * 64 + j * 16 + ml;
                Bv[j] = *(const v2f*)(as + n * 68 + kk + khalf);
            }
            #pragma unroll
            for (int i = 0; i < 2; ++i)
                #pragma unroll
                for (int j = 0; j < 4; ++j)
                    acc[i][j] = wmma_f32x4(A[i], Bv[j], acc[i][j]);
        }
    }

    // epilogue: scale, mask, store S
    const int mh = (lane >> 4) * 8;
    const int nl = lane & 15;
    #pragma unroll
    for (int j = 0; j < 4; ++j) {
        int n = aBase + wn * 64 + j * 16 + nl;
        int amv = amask[b * LA_ + n];
        #pragma unroll
        for (int i = 0; i < 2; ++i) {
            int mB = qBase + wm * 32 + i * 16 + mh;
            #pragma unroll
            for (int r = 0; r < 8; ++r) {
                int m = mB + r;
                int qmv = qmask[b * LQ_ + m];
                float v = acc[i][j][r] * T;
                v = (qmv && amv) ? v : NEG_INF_;
                S[((size_t)b * LQ_ + m) * LA_ + n] = v;
            }
        }
    }
}

// ---------------------------------------------------------------------------
// Kernel 2a: per-row (axis=2, over La) max and 1/sum(exp)
// ---------------------------------------------------------------------------
__global__ __launch_bounds__(256) void row_stats_kernel(
    const float* __restrict__ S, float* __restrict__ rowM, float* __restrict__ rowInv)
{
    const int row = blockIdx.x;                 // 0 .. B*LQ-1
    const float* s = S + (size_t)row * LA_;
    const int tid = threadIdx.x;

    float vals[4];
    float m = -INFINITY;
    #pragma unroll
    for (int i = 0; i < 4; ++i) { vals[i] = s[tid + i * 256]; m = fmaxf(m, vals[i]); }

    __shared__ float red[256];
    red[tid] = m; __syncthreads();
    for (int sft = 128; sft > 0; sft >>= 1) {
        if (tid < sft) red[tid] = fmaxf(red[tid], red[tid + sft]);
        __syncthreads();
    }
    m = red[0]; __syncthreads();

    float sum = 0.f;
    #pragma unroll
    for (int i = 0; i < 4; ++i) sum += __expf(vals[i] - m);
    red[tid] = sum; __syncthreads();
    for (int sft = 128; sft > 0; sft >>= 1) {
        if (tid < sft) red[tid] += red[tid + sft];
        __syncthreads();
    }
    if (tid == 0) { rowM[row] = m; rowInv[row] = 1.f / red[0]; }
}

// ---------------------------------------------------------------------------
// Kernel 2b: per-column (axis=1, over Lq) online-softmax stats.
// Block covers 32 columns x 1024 rows; lane = column, wave = 128-row strip.
// ---------------------------------------------------------------------------
__global__ __launch_bounds__(256) void col_stats_kernel(
    const float* __restrict__ S, float* __restrict__ colM, float* __restrict__ colInv)
{
    const int b     = blockIdx.y;
    const int aBase = blockIdx.x * 32;
    const int tid   = threadIdx.x;
    const int lane  = tid & 31;
    const int wave  = tid >> 5;

    const float* s = S + ((size_t)b * LQ_ + wave * 128) * LA_ + aBase + lane;
    float m = -INFINITY, sum = 0.f;
    for (int r = 0; r < 128; ++r) {
        float x  = s[(size_t)r * LA_];
        float mn = fmaxf(m, x);
        sum = sum * __expf(m - mn) + __expf(x - mn);
        m = mn;
    }

    __shared__ float mred[8][32], sred[8][32];
    mred[wave][lane] = m; sred[wave][lane] = sum;
    __syncthreads();
    if (wave == 0) {
        float M = mred[0][lane], Sm = sred[0][lane];
        #pragma unroll
        for (int w = 1; w < 8; ++w) {
            float mw = mred[w][lane], sw = sred[w][lane];
            float mn = fmaxf(M, mw);
            Sm = Sm * __expf(M - mn) + sw * __expf(mw - mn);
            M = mn;
        }
        colM[b * LA_ + aBase + lane]   = M;
        colInv[b * LA_ + aBase + lane] = 1.f / Sm;
    }
}

// ---------------------------------------------------------------------------
// Kernel 3: q_s = softmax_a(S) @ a  -> out_q[b, q, 512 + h]
// A operand = exp(S - rowM) (M=q, K=a), B operand = a chunk (K=a, N=h).
// ---------------------------------------------------------------------------
__global__ __launch_bounds__(256) void qs_agg_kernel(
    const float* __restrict__ S, const float* __restrict__ a,
    const float* __restrict__ rowM, const float* __restrict__ rowInv,
    float* __restrict__ outq)
{
    const int b     = blockIdx.z;
    const int qBase = blockIdx.y * 128;
    const int hBase = blockIdx.x * 128;
    const int tid   = threadIdx.x;
    const int lane  = tid & 31;
    const int wave  = tid >> 5;
    const int wm    = wave & 3;
    const int wn    = wave >> 2;

    __shared__ __align__(16) float Ps[128 * 68];   // exp(S) tile, [q][k]
    __shared__ __align__(16) float Bs[64 * 132];   // a chunk,    [k][h]

    v8f acc[2][4];
    #pragma unroll
    for (int i = 0; i < 2; ++i)
        #pragma unroll
        for (int j = 0; j < 4; ++j) acc[i][j] = {};

    for (int k0 = 0; k0 < LA_; k0 += 64) {
        __syncthreads();
        #pragma unroll
        for (int i = 0; i < 8; ++i) {
            int idx = tid + i * 256;       // 2048 float4 = 128 x 16
            int r = idx >> 4, c4 = idx & 15;
            float rm = rowM[b * LQ_ + qBase + r];
            float4 v = *(const float4*)(S + ((size_t)b * LQ_ + qBase + r) * LA_ + k0 + c4 * 4);
            float4 p;
            p.x = __expf(v.x - rm); p.y = __expf(v.y - rm);
            p.z = __expf(v.z - rm); p.w = __expf(v.w - rm);
            *(float4*)(Ps + r * 68 + c4 * 4) = p;
        }
        #pragma unroll
        for (int i = 0; i < 8; ++i) {
            int idx = tid + i * 256;       // 2048 float4 = 64 x 32
            int r = idx >> 5, c4 = idx & 31;
            float4 v = *(const float4*)(a + ((size_t)b * LA_ + k0 + r) * H_ + hBase + c4 * 4);
            *(float4*)(Bs + r * 132 + c4 * 4) = v;
        }
        __syncthreads();

        const int khalf = (lane >> 4) * 2;
        const int ml    = lane & 15;
        #pragma unroll
        for (int kk = 0; kk < 64; kk += 4) {
            v2f A[2], Bv[4];
            #pragma unroll
            for (int i = 0; i < 2; ++i) {
                int m = wm * 32 + i * 16 + ml;
                A[i] = *(const v2f*)(Ps + m * 68 + kk + khalf);
            }
            #pragma unroll
            for (int j = 0; j < 4; ++j) {
                int n = wn * 64 + j * 16 + ml;
                v2f bv;
                bv.x = Bs[(kk + khalf + 0) * 132 + n];
                bv.y = Bs[(kk + khalf + 1) * 132 + n];
                Bv[j] = bv;
            }
            #pragma unroll
            for (int i = 0; i < 2; ++i)
                #pragma unroll
                for (int j = 0; j < 4; ++j)
                    acc[i][j] = wmma_f32x4(A[i], Bv[j], acc[i][j]);
        }
    }

    const int mh = (lane >> 4) * 8;
    const int nl = lane & 15;
    #pragma unroll
    for (int j = 0; j < 4; ++j) {
        int h = hBase + wn * 64 + j * 16 + nl;
        #pragma unroll
        for (int i = 0; i < 2; ++i) {
            int mB = qBase + wm * 32 + i * 16 + mh;
            #pragma unroll
            for (int r = 0; r < 8; ++r) {
                int m = mB + r;
                float ri = rowInv[b * LQ_ + m];
                outq[((size_t)b * LQ_ + m) * (2 * H_) + H_ + h] = acc[i][j][r] * ri;
            }
        }
    }
}

// ---------------------------------------------------------------------------
// Kernel 4: a_s = softmax_q(S)^T @ q -> out_a[b, a, 512 + h]
// A operand = exp(S - colM) transposed (M=a, K=q), B operand = q chunk.
// ---------------------------------------------------------------------------
__global__ __launch_bounds__(256) void as_agg_kernel(
    const float* __restrict__ S, const float* __restrict__ qmat,
    const float* __restrict__ colM, const float* __restrict__ colInv,
    float* __restrict__ outa)
{
    const int b     = blockIdx.z;
    const int aBase = blockIdx.y * 128;   // output rows (a index)
    const int hBase = blockIdx.x * 128;
    const int tid   = threadIdx.x;
    const int lane  = tid & 31;
    const int wave  = tid >> 5;
    const int wm    = wave & 3;
    const int wn    = wave >> 2;

    __shared__ __align__(16) float Pt[128 * 68];   // exp(S)^T tile, [a][k]
    __shared__ __align__(16) float Bs[64 * 132];   // q chunk,       [k][h]

    v8f acc[2][4];
    #pragma unroll
    for (int i = 0; i < 2; ++i)
        #pragma unroll
        for (int j = 0; j < 4; ++j) acc[i][j] = {};

    for (int k0 = 0; k0 < LQ_; k0 += 64) {
        __syncthreads();
        // S rows k0..k0+63 (q), cols aBase..+127 (a); exp + transpose into Pt
        #pragma unroll
        for (int i = 0; i < 8; ++i) {
            int idx = tid + i * 256;       // 2048 float4 = 64 x 32
            int r = idx >> 5, c4 = idx & 31;
            int acol = aBase + c4 * 4;
            float4 v  = *(const float4*)(S + ((size_t)b * LQ_ + k0 + r) * LA_ + acol);
            float4 cm = *(const float4*)(colM + b * LA_ + acol);
            Pt[(c4 * 4 + 0) * 68 + r] = __expf(v.x - cm.x);
            Pt[(c4 * 4 + 1) * 68 + r] = __expf(v.y - cm.y);
            Pt[(c4 * 4 + 2) * 68 + r] = __expf(v.z - cm.z);
            Pt[(c4 * 4 + 3) * 68 + r] = __expf(v.w - cm.w);
        }
        #pragma unroll
        for (int i = 0; i < 8; ++i) {
            int idx = tid + i * 256;
            int r = idx >> 5, c4 = idx & 31;
            float4 v = *(const float4*)(qmat + ((size_t)b * LQ_ + k0 + r) * H_ + hBase + c4 * 4);
            *(float4*)(Bs + r * 132 + c4 * 4) = v;
        }
        __syncthreads();

        const int khalf = (lane >> 4) * 2;
        const int ml    = lane & 15;
        #pragma unroll
        for (int kk = 0; kk < 64; kk += 4) {
            v2f A[2], Bv[4];
            #pragma unroll
            for (int i = 0; i < 2; ++i) {
                int m = wm * 32 + i * 16 + ml;
                A[i] = *(const v2f*)(Pt + m * 68 + kk + khalf);
            }
            #pragma unroll
            for (int j = 0; j < 4; ++j) {
                int n = wn * 64 + j * 16 + ml;
                v2f bv;
                bv.x = Bs[(kk + khalf + 0) * 132 + n];
                bv.y = Bs[(kk + khalf + 1) * 132 + n];
                Bv[j] = bv;
            }
            #pragma unroll
            for (int i = 0; i < 2; ++i)
                #pragma unroll
                for (int j = 0; j < 4; ++j)
                    acc[i][j] = wmma_f32x4(A[i], Bv[j], acc[i][j]);
        }
    }

    const int mh = (lane >> 4) * 8;
    const int nl = lane & 15;
    #pragma unroll
    for (int j = 0; j < 4; ++j) {
        int h = hBase + wn * 64 + j * 16 + nl;
        #pragma unroll
        for (int i = 0; i < 2; ++i) {
            int mB = aBase + wm * 32 + i * 16 + mh;
            #pragma unroll
            for (int r = 0; r < 8; ++r) {
                int m = mB + r;
                float ci = colInv[b * LA_ + m];
                outa[((size_t)b * LA_ + m) * (2 * H_) + H_ + h] = acc[i][j][r] * ci;
            }
        }
    }
}

// ---------------------------------------------------------------------------
// Kernel 5: copy q -> out_q[..., :512], a -> out_a[..., :512]  (float4)
// ---------------------------------------------------------------------------
__global__ __launch_bounds__(256) void copy_concat_kernel(
    const float* __restrict__ q, const float* __restrict__ a, float* __restrict__ out)
{
    const size_t n4 = (size_t)B_ * LQ_ * (H_ / 4);     // float4 count per tensor
    size_t i = (size_t)blockIdx.x * blockDim.x + threadIdx.x;
    if (i < n4) {
        size_t row = i >> 7;         // / (512/4)
        size_t col = i & 127;
        const float4* q4 = (const float4*)q;
        const float4* a4 = (const float4*)a;
        float4* o4 = (float4*)out;
        o4[row * 256 + col] = q4[i];                                    // out_q low half
        o4[(size_t)B_ * LQ_ * 256 + row * 256 + col] = a4[i];           // out_a low half
    }
}

// ---------------------------------------------------------------------------
extern "C" void kernel_launch(void* const* d_in, const int* in_sizes, int n_in,
                              void* d_out, int out_size, void* d_ws, size_t ws_size,
                              hipStream_t stream) {
    (void)in_sizes; (void)n_in; (void)out_size; (void)ws_size;

    const float* q    = (const float*)d_in[0];
    const float* a    = (const float*)d_in[1];
    const int*   qm   = (const int*)d_in[2];
    const int*   am   = (const int*)d_in[3];
    const float* temp = (const float*)d_in[4];

    float* out  = (float*)d_out;
    float* outa = out + (size_t)B_ * LQ_ * (2 * H_);

    float* S       = (float*)d_ws;                       // 128 MB, L2-resident
    size_t Selems  = (size_t)B_ * LQ_ * LA_;
    float* rowM    = S + Selems;
    float* rowInv  = rowM + (size_t)B_ * LQ_;
    float* colM    = rowInv + (size_t)B_ * LQ_;
    float* colInv  = colM + (size_t)B_ * LA_;

    // 1) scores + mask
    score_kernel<<<dim3(LA_ / 128, LQ_ / 128, B_), 256, 0, stream>>>(q, a, qm, am, temp, S);
    // 2) softmax statistics for both axes
    row_stats_kernel<<<dim3(B_ * LQ_), 256, 0, stream>>>(S, rowM, rowInv);
    col_stats_kernel<<<dim3(LA_ / 32, B_), 256, 0, stream>>>(S, colM, colInv);
    // 3) aggregations, written directly into the high halves of the outputs
    qs_agg_kernel<<<dim3(H_ / 128, LQ_ / 128, B_), 256, 0, stream>>>(S, a, rowM, rowInv, out);
    as_agg_kernel<<<dim3(H_ / 128, LA_ / 128, B_), 256, 0, stream>>>(S, q, colM, colInv, outa);
    // 4) pass-through halves of the concatenation
    size_t n4 = (size_t)B_ * LQ_ * (H_ / 4);
    copy_concat_kernel<<<dim3((unsigned)((n4 + 255) / 256)), 256, 0, stream>>>(q, a, out);
}